// ShiftedWindowMSA_64819646431313
// MI455X (gfx1250) — compile-verified
//
#include <hip/hip_runtime.h>

// ---------------- problem constants ----------------
#define HW_    56          // spatial h=w
#define CTOT   96          // channels
#define HEADS  3
#define EDIM   32
#define WIN    7
#define TOK    49          // tokens per window
#define TPAD   64          // padded token count (4 x 16 WMMA tiles)
#define NQKV   288         // 3*C
#define WQS    292         // LDS row stride for staged w_qkv (16B-aligned rows)

typedef __attribute__((ext_vector_type(2))) float v2f;
typedef __attribute__((ext_vector_type(8))) float v8f;

#define WMMA4(a, b, c) \
  __builtin_amdgcn_wmma_f32_16x16x4_f32(false, (a), false, (b), (short)0, (c), false, false)

__global__ __launch_bounds__(256) void swin_msa_fused(
    const float* __restrict__ x, const float* __restrict__ w_qkv,
    const float* __restrict__ b_qkv, const float* __restrict__ w_proj,
    const float* __restrict__ b_proj, const float* __restrict__ rel_bias,
    float* __restrict__ out)
{
  // ~283 KB of LDS per workgroup — CDNA5 WGP has 320 KB.
  __shared__ float xs[TPAD][100];            // x window tile (rows >=49 zeroed)
  __shared__ float wq[CTOT][WQS];            // staged w_qkv [96][288] (pad 292)
  __shared__ float qs[HEADS][TPAD][33];      // Q per head, padded stride 33
  __shared__ float ks[HEADS][TPAD][33];      // K per head
  __shared__ float vs[HEADS][TPAD][33];      // V per head
  __shared__ float sc[HEADS][TPAD][65];      // scores -> attn probs
  __shared__ float ow[TPAD][100];            // attention output window [64][96]

  const int tid  = threadIdx.x;
  const int wid  = tid >> 5;        // wave id 0..7
  const int lane = tid & 31;
  const int ln16 = lane & 15;       // M (A/C) or N (B/C) index
  const int kh   = lane >> 4;       // K half-select for A/B fragments
  const int rofs = kh * 8;          // C/D row offset for upper half-wave

  const int blk = blockIdx.x;       // = b*64 + window
  const int b   = blk >> 6;
  const int win = blk & 63;
  const int wy  = win >> 3;
  const int wx  = win & 7;
  const long xbase = (long)b * (HW_ * HW_) * CTOT;

  // ---- phase 0a: stage w_qkv [96][288] into LDS with float4 loads
  for (int idx = tid; idx < CTOT * (NQKV / 4); idx += 256) {
    int r  = idx / (NQKV / 4);
    int c4 = idx - r * (NQKV / 4);
    float4 v = *(const float4*)(w_qkv + r * NQKV + c4 * 4);
    *(float4*)&wq[r][c4 * 4] = v;
  }
  // ---- phase 0b: stage the 49x96 window of x into LDS (zero-pad to 64 rows)
  for (int idx = tid; idx < TPAD * (CTOT / 4); idx += 256) {
    int r  = idx / (CTOT / 4);
    int c4 = idx - r * (CTOT / 4);
    float4 v = make_float4(0.f, 0.f, 0.f, 0.f);
    if (r < TOK) {
      int m1 = r / WIN, m2 = r % WIN;
      int grow = (wy * WIN + m1) * HW_ + (wx * WIN + m2);
      v = *(const float4*)(x + xbase + (long)grow * CTOT + c4 * 4);
    }
    *(float4*)&xs[r][c4 * 4] = v;
  }
  __syncthreads();

  // ---- phase 1: qkv = xs @ w_qkv + b_qkv, scatter into Q/K/V (deinterleave H,E,3)
  for (int tt = wid; tt < 4 * 18; tt += 8) {
    const int row0 = (tt / 18) * 16;
    const int col0 = (tt % 18) * 16;
    v8f acc = {};
#pragma unroll
    for (int kk = 0; kk < CTOT; kk += 4) {
      v2f a, bf;
      a.x  = xs[row0 + ln16][kk + kh * 2 + 0];
      a.y  = xs[row0 + ln16][kk + kh * 2 + 1];
      bf.x = wq[kk + kh * 2 + 0][col0 + ln16];
      bf.y = wq[kk + kh * 2 + 1][col0 + ln16];
      acc = WMMA4(a, bf, acc);
    }
#pragma unroll
    for (int j = 0; j < 8; ++j) {
      int row = row0 + j + rofs;
      int col = col0 + ln16;
      float val = acc[j] + b_qkv[col];
      int head = col / CTOT;
      int r96  = col - head * CTOT;
      int e    = r96 / 3;
      int kqv  = r96 - 3 * e;
      if (kqv == 0)      qs[head][row][e] = val;
      else if (kqv == 1) ks[head][row][e] = val;
      else               vs[head][row][e] = val;
    }
  }
  __syncthreads();

  // ---- phase 2: scores = Q @ K^T * scale + rel_bias (mask padded keys)
  const float scale = 0.17677669529663687f;  // 1/sqrt(32)
  for (int tt = wid; tt < HEADS * 16; tt += 8) {
    const int head = tt / 16;
    const int q0   = ((tt % 16) / 4) * 16;
    const int j0   = (tt % 4) * 16;
    v8f acc = {};
#pragma unroll
    for (int kk = 0; kk < EDIM; kk += 4) {
      v2f a, bf;
      a.x  = qs[head][q0 + ln16][kk + kh * 2 + 0];
      a.y  = qs[head][q0 + ln16][kk + kh * 2 + 1];
      bf.x = ks[head][j0 + ln16][kk + kh * 2 + 0];   // B[k][n] = K[j0+n][k]
      bf.y = ks[head][j0 + ln16][kk + kh * 2 + 1];
      acc = WMMA4(a, bf, acc);
    }
#pragma unroll
    for (int j = 0; j < 8; ++j) {
      int row = q0 + j + rofs;
      int col = j0 + ln16;
      float v;
      if (col < TOK) {
        float bias = (row < TOK) ? rel_bias[row * TOK + col] : 0.f;
        v = acc[j] * scale + bias;
      } else {
        v = -1e30f;                 // padded key -> zero weight after softmax
      }
      sc[head][row][col] = v;
    }
  }
  __syncthreads();

  // ---- phase 3: softmax over keys (wave-parallel, shfl reductions)
  for (int task = wid; task < HEADS * TOK; task += 8) {
    const int head = task / TOK;
    const int r    = task - head * TOK;
    float v0 = sc[head][r][lane];
    float v1 = sc[head][r][lane + 32];
    float mx = fmaxf(v0, v1);
    for (int off = 16; off > 0; off >>= 1) mx = fmaxf(mx, __shfl_xor(mx, off, 32));
    float e0 = __expf(v0 - mx);
    float e1 = __expf(v1 - mx);
    float s = e0 + e1;
    for (int off = 16; off > 0; off >>= 1) s += __shfl_xor(s, off, 32);
    float inv = 1.f / s;
    sc[head][r][lane]      = e0 * inv;
    sc[head][r][lane + 32] = e1 * inv;
  }
  __syncthreads();

  // ---- phase 4: out = P @ V  -> window output [64][96]
  for (int tt = wid; tt < HEADS * 4 * 2; tt += 8) {
    const int head = tt / 8;
    const int q0   = ((tt % 8) / 2) * 16;
    const int e0   = (tt % 2) * 16;
    v8f acc = {};
#pragma unroll
    for (int kk = 0; kk < TPAD; kk += 4) {
      v2f a, bf;
      a.x  = sc[head][q0 + ln16][kk + kh * 2 + 0];
      a.y  = sc[head][q0 + ln16][kk + kh * 2 + 1];
      bf.x = vs[head][kk + kh * 2 + 0][e0 + ln16];   // B[k][n] = V[k][e0+n]
      bf.y = vs[head][kk + kh * 2 + 1][e0 + ln16];
      acc = WMMA4(a, bf, acc);
    }
#pragma unroll
    for (int j = 0; j < 8; ++j) {
      int row = q0 + j + rofs;
      int col = e0 + ln16;
      ow[row][head * EDIM + col] = acc[j];
    }
  }
  __syncthreads();

  // ---- phase 5: final projection + bias, store the 49 real rows
  //      (w_proj is 36 KB -> lives in the 64 KB WGP$, read from global)
  for (int tt = wid; tt < 4 * 6; tt += 8) {
    const int row0 = (tt / 6) * 16;
    const int col0 = (tt % 6) * 16;
    v8f acc = {};
#pragma unroll
    for (int kk = 0; kk < CTOT; kk += 4) {
      v2f a, bf;
      a.x  = ow[row0 + ln16][kk + kh * 2 + 0];
      a.y  = ow[row0 + ln16][kk + kh * 2 + 1];
      bf.x = w_proj[(kk + kh * 2 + 0) * CTOT + col0 + ln16];
      bf.y = w_proj[(kk + kh * 2 + 1) * CTOT + col0 + ln16];
      acc = WMMA4(a, bf, acc);
    }
#pragma unroll
    for (int j = 0; j < 8; ++j) {
      int row = row0 + j + rofs;
      if (row < TOK) {
        int col = col0 + ln16;
        int m1 = row / WIN, m2 = row % WIN;
        long grow = (long)b * (HW_ * HW_) +
                    (long)((wy * WIN + m1) * HW_ + (wx * WIN + m2));
        out[grow * CTOT + col] = acc[j] + b_proj[col];
      }
    }
  }
}

extern "C" void kernel_launch(void* const* d_in, const int* in_sizes, int n_in,
                              void* d_out, int out_size, void* d_ws, size_t ws_size,
                              hipStream_t stream) {
  const float* x        = (const float*)d_in[0];
  const float* w_qkv    = (const float*)d_in[1];
  const float* b_qkv    = (const float*)d_in[2];
  const float* w_proj   = (const float*)d_in[3];
  const float* b_proj   = (const float*)d_in[4];
  const float* rel_bias = (const float*)d_in[5];
  float* out = (float*)d_out;
  (void)in_sizes; (void)n_in; (void)out_size; (void)d_ws; (void)ws_size;

  // one workgroup per (batch, window): 64 batches * 64 windows
  swin_msa_fused<<<64 * 64, 256, 0, stream>>>(x, w_qkv, b_qkv, w_proj, b_proj,
                                              rel_bias, out);
}